// Att_cov_65704409694828
// MI455X (gfx1250) — compile-verified
//
#include <hip/hip_runtime.h>
#include <math.h>

typedef __attribute__((ext_vector_type(2))) float v2f;
typedef __attribute__((ext_vector_type(8))) float v8f;

// ---------------------------------------------------------------------------
// K1: h = x @ W_gcn ; deg = 1 (self loop) ; acc = 0
// ---------------------------------------------------------------------------
__global__ void k_node_init(const float* __restrict__ x,
                            const float* __restrict__ Wg,
                            float* __restrict__ h,
                            float* __restrict__ deg,
                            float* __restrict__ acc, int N) {
  int n = blockIdx.x * blockDim.x + threadIdx.x;
  if (n >= N) return;
  const float4* xr = (const float4*)(x + (long)n * 64);
  const float4* w4 = (const float4*)Wg;
  float s = 0.0f;
#pragma unroll
  for (int i = 0; i < 16; ++i) {
    float4 a = xr[i], b = w4[i];
    s += a.x * b.x + a.y * b.y + a.z * b.z + a.w * b.w;
  }
  h[n] = s;
  deg[n] = 1.0f;
  acc[n] = 0.0f;
}

// ---------------------------------------------------------------------------
// K2: degree accumulation over destination nodes
// ---------------------------------------------------------------------------
__global__ void k_degree(const int* __restrict__ col, float* __restrict__ deg,
                         int E) {
  int e = blockIdx.x * blockDim.x + threadIdx.x;
  if (e < E) atomicAdd(&deg[col[e]], 1.0f);
}

// ---------------------------------------------------------------------------
// K3: edge attention via V_WMMA_F32_16X16X4_F32 (16 edges per wave, K=128
// in 32 chained WMMAs, full fp32) + fused GCN scatter-add.
//
// A (16x4 f32): lanes 0-15 hold K0,K1 ; lanes 16-31 hold K2,K3 ; M = lane%16.
// B (4x16 f32): same K striping with N = lane%16; we replicate w[k] across
// all N columns, so every column of D equals the 16 edge dot-products.
// D (16x16 f32): VGPR j: lanes 0-15 -> M=j, lanes 16-31 -> M=j+8; lanes with
// N==0 (lanes 0 and 16) write out edges [0..7] / [8..15] of the tile.
// ---------------------------------------------------------------------------
__global__ void __launch_bounds__(256) k_edge(
    const float* __restrict__ x, const int* __restrict__ row,
    const int* __restrict__ col, const float* __restrict__ We,
    const float* __restrict__ be, const float* __restrict__ h,
    const float* __restrict__ deg, float* __restrict__ acc,
    float* __restrict__ out_m, float* __restrict__ out_s, int E) {
  int wave = (blockIdx.x * blockDim.x + threadIdx.x) >> 5;
  int lane = threadIdx.x & 31;
  int m = lane & 15;
  int hi = lane >> 4;

  long tile = (long)wave * 16;
  if (tile >= E) return;  // wave-uniform: EXEC stays all-ones inside

  int e = (int)tile + m;
  int ec = (e < E) ? e : (E - 1);  // clamp (E%16==0 here, but stay safe)
  int r = row[ec];
  int c = col[ec];

  const float* xr = x + (long)r * 64 + 2 * hi;
  const float* xc = x + (long)c * 64 + 2 * hi;
  const float* w0 = We + 2 * hi;        // first 64 weights (row half)
  const float* w1 = We + 64 + 2 * hi;   // second 64 weights (col half)

  v8f c8 = {0.f, 0.f, 0.f, 0.f, 0.f, 0.f, 0.f, 0.f};
#pragma unroll
  for (int kb = 0; kb < 64; kb += 4) {
    v2f a, b;
    a.x = xr[kb]; a.y = xr[kb + 1];
    b.x = w0[kb]; b.y = w0[kb + 1];
    c8 = __builtin_amdgcn_wmma_f32_16x16x4_f32(false, a, false, b, (short)0,
                                               c8, false, false);
  }
#pragma unroll
  for (int kb = 0; kb < 64; kb += 4) {
    v2f a, b;
    a.x = xc[kb]; a.y = xc[kb + 1];
    b.x = w1[kb]; b.y = w1[kb + 1];
    c8 = __builtin_amdgcn_wmma_f32_16x16x4_f32(false, a, false, b, (short)0,
                                               c8, false, false);
  }

  // write-out: lanes 0 and 16 own column N==0
  if (m == 0) {
    float bias = be[0];
    long base = tile + (long)hi * 8;
#pragma unroll
    for (int j = 0; j < 8; ++j) {
      long ee = base + j;
      if (ee < E) {
        float z = c8[j] + bias;
        float sg = 1.0f / (1.0f + expf(-z));
        out_m[ee] = sg;
        out_s[ee] = 1.0f - sg;
      }
    }
  }

  // fused GCN scatter: one lane (hi==0) per edge
  if (hi == 0 && e < E) {
    float contrib = rsqrtf(deg[r]) * rsqrtf(deg[c]) * h[r];
    atomicAdd(&acc[c], contrib);
  }
}

// ---------------------------------------------------------------------------
// K4: node_att = acc + h/deg (self-loop norm = dinv^2) + b_gcn  (in place)
// ---------------------------------------------------------------------------
__global__ void k_node_final(float* __restrict__ acc,
                             const float* __restrict__ h,
                             const float* __restrict__ deg,
                             const float* __restrict__ bg, int N) {
  int n = blockIdx.x * blockDim.x + threadIdx.x;
  if (n >= N) return;
  acc[n] = acc[n] + h[n] / deg[n] + bg[0];
}

// ---------------------------------------------------------------------------
// K5: per-graph segment softmax, twice (m = softmax(att), s = softmax(1-m)).
// One block per graph; nodes/graph (=100 here) buffered in LDS.
// ---------------------------------------------------------------------------
#define SM_THREADS 128
#define SM_MAXSEG 1024

__device__ __forceinline__ float blockMax(float v, float* s) {
  int t = threadIdx.x;
  s[t] = v;
  __syncthreads();
  for (int st = SM_THREADS >> 1; st > 0; st >>= 1) {
    if (t < st) s[t] = fmaxf(s[t], s[t + st]);
    __syncthreads();
  }
  float r = s[0];
  __syncthreads();
  return r;
}

__device__ __forceinline__ float blockSum(float v, float* s) {
  int t = threadIdx.x;
  s[t] = v;
  __syncthreads();
  for (int st = SM_THREADS >> 1; st > 0; st >>= 1) {
    if (t < st) s[t] = s[t] + s[t + st];
    __syncthreads();
  }
  float r = s[0];
  __syncthreads();
  return r;
}

__global__ void __launch_bounds__(SM_THREADS) k_softmax(
    const float* __restrict__ att, const int* __restrict__ split, int G,
    float* __restrict__ out_m, float* __restrict__ out_s) {
  __shared__ float red[SM_THREADS];
  __shared__ float buf[SM_MAXSEG];
  int g = blockIdx.x;
  int t = threadIdx.x;

  int off = 0;
  for (int i = 0; i < g; ++i) off += split[i];  // uniform scalar walk (G=500)
  int cnt = split[g];

  // ---- pass 1: m = softmax(att) ----
  float mx = -3.402823e38f;
  for (int i = t; i < cnt; i += SM_THREADS) mx = fmaxf(mx, att[off + i]);
  mx = blockMax(mx, red);

  float sum = 0.0f;
  for (int i = t; i < cnt; i += SM_THREADS) {
    float ev = expf(att[off + i] - mx);
    buf[i] = ev;
    sum += ev;
  }
  sum = blockSum(sum, red);
  float inv = 1.0f / sum;
  for (int i = t; i < cnt; i += SM_THREADS) {
    float mval = buf[i] * inv;
    buf[i] = mval;  // keep m for pass 2
    out_m[off + i] = mval;
  }
  __syncthreads();

  // ---- pass 2: s = softmax(1 - m) ----
  float mx2 = -3.402823e38f;
  for (int i = t; i < cnt; i += SM_THREADS) mx2 = fmaxf(mx2, 1.0f - buf[i]);
  mx2 = blockMax(mx2, red);

  float sum2 = 0.0f;
  for (int i = t; i < cnt; i += SM_THREADS)
    sum2 += expf((1.0f - buf[i]) - mx2);
  sum2 = blockSum(sum2, red);
  float inv2 = 1.0f / sum2;
  for (int i = t; i < cnt; i += SM_THREADS)
    out_s[off + i] = expf((1.0f - buf[i]) - mx2) * inv2;
}

// ---------------------------------------------------------------------------
// launch
// ---------------------------------------------------------------------------
extern "C" void kernel_launch(void* const* d_in, const int* in_sizes, int n_in,
                              void* d_out, int out_size, void* d_ws,
                              size_t ws_size, hipStream_t stream) {
  const float* x = (const float*)d_in[0];
  const int* ei = (const int*)d_in[1];
  const int* split = (const int*)d_in[2];
  const float* We = (const float*)d_in[3];
  const float* be = (const float*)d_in[4];
  const float* Wg = (const float*)d_in[5];
  const float* bg = (const float*)d_in[6];

  const int N = in_sizes[0] / 64;
  const int E = in_sizes[1] / 2;
  const int G = in_sizes[2];

  const int* row = ei;
  const int* col = ei + E;

  float* h = (float*)d_ws;
  float* deg = h + N;
  float* acc = deg + N;

  float* out = (float*)d_out;
  float* out_em = out;                 // edge_att_m [E]
  float* out_es = out + E;             // edge_att_s [E]
  float* out_nm = out + 2 * (long)E;   // node_att_m [N]
  float* out_ns = out_nm + N;          // node_att_s [N]

  // K1: h, deg=1, acc=0
  k_node_init<<<(N + 255) / 256, 256, 0, stream>>>(x, Wg, h, deg, acc, N);
  // K2: degree over col
  k_degree<<<(E + 255) / 256, 256, 0, stream>>>(col, deg, E);
  // K3: WMMA edge attention + fused GCN scatter
  {
    long waves = ((long)E + 15) / 16;
    long threads = waves * 32;
    int blocks = (int)((threads + 255) / 256);
    k_edge<<<blocks, 256, 0, stream>>>(x, row, col, We, be, h, deg, acc,
                                       out_em, out_es, E);
  }
  // K4: finalize node attention (in acc)
  k_node_final<<<(N + 255) / 256, 256, 0, stream>>>(acc, h, deg, bg, N);
  // K5: per-graph double softmax
  k_softmax<<<G, SM_THREADS, 0, stream>>>(acc, split, G, out_nm, out_ns);
}